// GenotypeNet_1065151889591
// MI455X (gfx1250) — compile-verified
//
#include <hip/hip_runtime.h>
#include <cstdint>

// ---- Problem constants (match reference setup_inputs) ----
#define N_IN   512
#define LAYERS 5
#define M      2048
#define FAN    32
#define BATCH  1024
#define NTOT   (N_IN + LAYERS * M)   // 10752
#define EDGES  (M * FAN)             // 65536
#define CHUNK  256                   // batch elements per block
#define NCHUNK (BATCH / CHUNK)       // 4
#define NPAIRS 8                     // node-pairs per TDM block (16 nodes)

typedef uint32_t v4u __attribute__((ext_vector_type(4)));
typedef uint32_t v8u __attribute__((ext_vector_type(8)));

// ---------------------------------------------------------------------------
// Tiled transpose: x[B][N_IN] (row-major) -> valsT[node][batch]
// ---------------------------------------------------------------------------
__global__ void __launch_bounds__(256) k_transpose_in(const float* __restrict__ x,
                                                      float* __restrict__ valsT) {
    __shared__ float tile[32][33];
    const int n0 = blockIdx.x * 32;
    const int b0 = blockIdx.y * 32;
    const int tx = threadIdx.x, ty = threadIdx.y;      // 32 x 8
    #pragma unroll
    for (int i = ty; i < 32; i += 8)
        tile[i][tx] = x[(size_t)(b0 + i) * N_IN + (n0 + tx)];
    __syncthreads();
    #pragma unroll
    for (int i = ty; i < 32; i += 8)
        valsT[(size_t)(n0 + i) * BATCH + (b0 + tx)] = tile[tx][i];
}

// ---------------------------------------------------------------------------
// Layer kernel, async-copy variant (ASYNCcnt path). One node per block.
// Used for layer 0 (sources confined to the 2 MB input region).
// ---------------------------------------------------------------------------
__global__ void __launch_bounds__(256) k_layer_async(const float* __restrict__ weights,
                                                     const float* __restrict__ biases,
                                                     const int*   __restrict__ src,
                                                     float*       __restrict__ valsT,
                                                     int l) {
    __shared__ __align__(16) float smem[FAN * CHUNK];   // 32 KB
    const int tid = threadIdx.x;
    const int m   = blockIdx.x;
    const int b0  = blockIdx.y * CHUNK;

    const int* __restrict__ srow = src + (size_t)l * EDGES + (size_t)m * FAN;
    const uint32_t smem_base = (uint32_t)(uintptr_t)&smem[0];  // LDS byte offset
    const int      c16  = tid & 63;
    const int      rsub = tid >> 6;
    const uint64_t base = (uint64_t)(uintptr_t)valsT;

    #pragma unroll
    for (int it = 0; it < 8; ++it) {
        const int      row   = it * 4 + rsub;
        const int      s     = srow[row];
        const uint32_t voff  = (uint32_t)((s * BATCH + b0) * 4) + (uint32_t)(c16 * 16);
        const uint32_t laddr = smem_base + (uint32_t)(row * CHUNK * 4 + c16 * 16);
        asm volatile("global_load_async_to_lds_b128 %0, %1, %2"
                     :: "v"(laddr), "v"(voff), "s"(base)
                     : "memory");
    }
    asm volatile("s_wait_asynccnt 0" ::: "memory");
    __syncthreads();

    const float* __restrict__ wrow = weights + (size_t)l * EDGES + (size_t)m * FAN;
    float acc = 0.0f;
    #pragma unroll
    for (int k = 0; k < FAN; ++k)
        acc = fmaf(smem[k * CHUNK + tid], wrow[k], acc);

    float v = acc + biases[(size_t)l * M + m];
    v = v > 0.0f ? v : 0.0f;
    valsT[(size_t)(N_IN + l * M + m) * BATCH + (b0 + tid)] = v;
}

// ---------------------------------------------------------------------------
// Issue one gather-mode tensor_load_to_lds: 8 rows x 256 f32 from
// valsT[row][b0..b0+255] into LDS at lds_base. Per-wave, EXEC-independent.
// ---------------------------------------------------------------------------
__device__ __forceinline__ void tdm_gather8(const int* __restrict__ idx8,
                                            uint32_t lds_base, uint64_t ga) {
    const uint32_t i0 = (uint32_t)idx8[0], i1 = (uint32_t)idx8[1];
    const uint32_t i2 = (uint32_t)idx8[2], i3 = (uint32_t)idx8[3];
    const uint32_t i4 = (uint32_t)idx8[4], i5 = (uint32_t)idx8[5];
    const uint32_t i6 = (uint32_t)idx8[6], i7 = (uint32_t)idx8[7];

    // D# group 0: count=1, gather_mode=1, 32-bit indices, type=2 ("image")
    v4u g0 = { 0xC0000001u,
               lds_base,
               (uint32_t)ga,
               ((uint32_t)(ga >> 32) & 0x01FFFFFFu) | (2u << 30) };
    // D# group 1: data_size=4B, tensor 1024 x 10752 (stride 1024), tile 256 x 8
    v8u g1 = { 2u << 16,                    // [17:16] data_size = 2 (4 bytes)
               (uint32_t)BATCH << 16,       // [63:48]  tensor_dim0 lo16
               (uint32_t)NTOT  << 16,       // [95:80]  tensor_dim1 lo16
               256u << 16,                  // [127:112] tile_dim0 = 256
               8u,                          // [143:128] tile_dim1 = #indices
               (uint32_t)BATCH,             // [191:160] tensor_dim0_stride lo32
               0u, 0u };
    // D# groups 2/3: eight 32-bit gather row indices
    v4u g2 = { i0, i1, i2, i3 };
    v4u g3 = { i4, i5, i6, i7 };

    asm volatile("tensor_load_to_lds %0, %1, %2, %3"
                 :: "s"(g0), "s"(g1), "s"(g2), "s"(g3)
                 : "memory");
}

// ---------------------------------------------------------------------------
// Pipelined TDM gather layer kernel (TENSORcnt path).
// Block = 16 nodes x 256-batch chunk, processed as 8 node-pairs with
// double-buffered LDS (2 x 64 KB). Each of the 8 waves issues ONE gather op
// (8 rows) per pair; the gather for pair p+1 is in flight while pair p is
// computed. TDM ops complete in-order per wave, so s_wait_tensorcnt 1 drains
// exactly the older gather. Buffer reuse is fenced by the trailing barrier.
// ---------------------------------------------------------------------------
__global__ void __launch_bounds__(256) k_layer_tdm(const float* __restrict__ weights,
                                                   const float* __restrict__ biases,
                                                   const int*   __restrict__ src,
                                                   float*       __restrict__ valsT,
                                                   int l) {
    __shared__ __align__(16) float smem[2][2 * FAN * CHUNK];  // 2 x 64 KB
    const int tid  = threadIdx.x;
    const int mb   = blockIdx.x * (2 * NPAIRS);   // first of 16 nodes
    const int b0   = blockIdx.y * CHUNK;
    const int wave = tid >> 5;                    // 0..7 (wave-uniform)

    const int* __restrict__ sbase = src + (size_t)l * EDGES + (size_t)mb * FAN;
    const uint64_t ga = (uint64_t)(uintptr_t)valsT + (uint64_t)b0 * 4u;
    const uint32_t lds0 = (uint32_t)(uintptr_t)&smem[0][0] + (uint32_t)wave * 8192u;
    const uint32_t lds1 = (uint32_t)(uintptr_t)&smem[1][0] + (uint32_t)wave * 8192u;

    // Prologue: gather pair 0 into buffer 0.
    tdm_gather8(sbase + wave * 8, lds0, ga);

    #pragma unroll
    for (int p = 0; p < NPAIRS; ++p) {
        if (p + 1 < NPAIRS) {
            // Prefetch pair p+1 into the other buffer, then drain pair p only.
            tdm_gather8(sbase + (size_t)(p + 1) * 64 + wave * 8,
                        ((p + 1) & 1) ? lds1 : lds0, ga);
            asm volatile("s_wait_tensorcnt 0x1" ::: "memory");
        } else {
            asm volatile("s_wait_tensorcnt 0x0" ::: "memory");
        }
        __syncthreads();   // all waves' slices of pair p visible

        const float* buf = &smem[p & 1][0];
        const int    m0  = mb + 2 * p;
        const float* __restrict__ wrow = weights + (size_t)l * EDGES + (size_t)m0 * FAN;

        float a0 = 0.0f, a1 = 0.0f;
        #pragma unroll
        for (int k = 0; k < FAN; ++k) {
            a0 = fmaf(buf[k * CHUNK + tid],         wrow[k],       a0);
            a1 = fmaf(buf[(k + FAN) * CHUNK + tid], wrow[k + FAN], a1);
        }
        float v0 = a0 + biases[(size_t)l * M + m0];
        float v1 = a1 + biases[(size_t)l * M + m0 + 1];
        v0 = v0 > 0.0f ? v0 : 0.0f;
        v1 = v1 > 0.0f ? v1 : 0.0f;
        valsT[(size_t)(N_IN + l * M + m0)     * BATCH + (b0 + tid)] = v0;
        valsT[(size_t)(N_IN + l * M + m0 + 1) * BATCH + (b0 + tid)] = v1;

        __syncthreads();   // everyone done reading buf before it is re-gathered
    }
}

// ---------------------------------------------------------------------------
// Tiled transpose of the last layer: valsT[node][batch] -> out[B][M]
// ---------------------------------------------------------------------------
__global__ void __launch_bounds__(256) k_transpose_out(const float* __restrict__ valsT,
                                                       float* __restrict__ out) {
    __shared__ float tile[32][33];
    const int m0 = blockIdx.x * 32;
    const int b0 = blockIdx.y * 32;
    const int tx = threadIdx.x, ty = threadIdx.y;      // 32 x 8
    const float* __restrict__ last = valsT + (size_t)(N_IN + (LAYERS - 1) * M) * BATCH;
    #pragma unroll
    for (int i = ty; i < 32; i += 8)
        tile[i][tx] = last[(size_t)(m0 + i) * BATCH + (b0 + tx)];
    __syncthreads();
    #pragma unroll
    for (int i = ty; i < 32; i += 8)
        out[(size_t)(b0 + i) * M + (m0 + tx)] = tile[tx][i];
}

// ---------------------------------------------------------------------------
// Launch: transpose-in, layer 0 via async-copy path, layers 1..4 via the
// pipelined TDM gather path, transpose-out. Workspace: 44 MB valsT.
// ---------------------------------------------------------------------------
extern "C" void kernel_launch(void* const* d_in, const int* in_sizes, int n_in,
                              void* d_out, int out_size, void* d_ws, size_t ws_size,
                              hipStream_t stream) {
    const float* x       = (const float*)d_in[0];   // [B, N_IN]
    const float* weights = (const float*)d_in[1];   // [L, E]
    const float* biases  = (const float*)d_in[2];   // [L, M]
    const int*   src     = (const int*)  d_in[3];   // [L, E]
    // d_in[4] (dst_idx) is implicit: repeat(arange(M), FAN)

    float* out   = (float*)d_out;                   // [B, M]
    float* valsT = (float*)d_ws;                    // [NTOT, BATCH]

    k_transpose_in<<<dim3(N_IN / 32, BATCH / 32), dim3(32, 8), 0, stream>>>(x, valsT);

    k_layer_async<<<dim3(M, NCHUNK), dim3(256), 0, stream>>>(weights, biases, src, valsT, 0);

    for (int l = 1; l < LAYERS; ++l)
        k_layer_tdm<<<dim3(M / (2 * NPAIRS), NCHUNK), dim3(256), 0, stream>>>(
            weights, biases, src, valsT, l);

    k_transpose_out<<<dim3(M / 32, BATCH / 32), dim3(32, 8), 0, stream>>>(valsT, out);
}